// GNN_67362267071067
// MI455X (gfx1250) — compile-verified
//
#include <hip/hip_runtime.h>
#include <math.h>

// Problem dims
#define BB    16
#define NNODE 12
#define TT    128
#define CCH   128
#define NE    144         // NNODE*NNODE
#define TC    16384       // TT*CCH
#define SZX   3145728     // BB*NNODE*TC
#define SZE   37748736    // BB*NE*TC
#define WP    132         // LDS weight row pitch (conflict-free for b64 frag reads)

typedef float v2f __attribute__((ext_vector_type(2)));
typedef float v8f __attribute__((ext_vector_type(8)));

// ---------------------------------------------------------------------------
// fp32 WMMA 16x16x4: D = A(16x4) * B(4x16) + C
// A layout: lane L holds row M=L%16, v2f = {K = k0+2*(L/16), K+1}
// B layout: lane L holds col N=L%16, v2f = {K = k0+2*(L/16), K+1}
// C/D layout: vgpr i -> M = i + 8*(L/16), N = L%16
// ---------------------------------------------------------------------------
__device__ __forceinline__ v8f wmma4(v2f a, v2f b, v8f c) {
  return __builtin_amdgcn_wmma_f32_16x16x4_f32(false, a, false, b, (short)0, c, false, false);
}

struct Strip { v8f acc[4]; };

// One wave: 16-row x 64-col strip of Y = X @ W^T, K=128.
// A streamed from global, B fragments served from LDS (pitch WP).
__device__ __forceinline__ void wmma_strip(const float* __restrict__ Arow0,
                                           const float* Blds,
                                           int n0, int lm, int half, Strip& s) {
  const float* a_ptr = Arow0 + (size_t)lm * CCH + 2 * half;
  const float* b0 = Blds + (n0 +  0 + lm) * WP + 2 * half;
  const float* b1 = Blds + (n0 + 16 + lm) * WP + 2 * half;
  const float* b2 = Blds + (n0 + 32 + lm) * WP + 2 * half;
  const float* b3 = Blds + (n0 + 48 + lm) * WP + 2 * half;
  s.acc[0] = v8f{}; s.acc[1] = v8f{}; s.acc[2] = v8f{}; s.acc[3] = v8f{};
#pragma unroll 8
  for (int k = 0; k < CCH; k += 4) {
    v2f a = *(const v2f*)(a_ptr + k);
    s.acc[0] = wmma4(a, *(const v2f*)(b0 + k), s.acc[0]);
    s.acc[1] = wmma4(a, *(const v2f*)(b1 + k), s.acc[1]);
    s.acc[2] = wmma4(a, *(const v2f*)(b2 + k), s.acc[2]);
    s.acc[3] = wmma4(a, *(const v2f*)(b3 + k), s.acc[3]);
  }
}

// Cooperative 128x128 weight stage into LDS with padded pitch (16B-aligned rows).
__device__ __forceinline__ void stage_weight(float* wlds, const float* __restrict__ W) {
  for (int u = threadIdx.x; u < CCH * (CCH / 4); u += 256) {
    int r  = u >> 5;        // row (32 float4 per row)
    int c4 = u & 31;
    float4 w = ((const float4*)W)[u];
    *(float4*)&wlds[r * WP + c4 * 4] = w;
  }
  __syncthreads();
}

// ---------------------------------------------------------------------------
// Kernel 1: y_w = x @ W_w.T  for W_w selected by blockIdx.y in {A,B,U,V}
// block = 256 (8 waves): wave -> {row-tile 0..3} x {col-block 0..1}
// grid = (384, 4); block stages its weight in LDS once.
// ---------------------------------------------------------------------------
__global__ __launch_bounds__(256) void k_xgemm(
    const float* __restrict__ x,
    const float* __restrict__ Wa, const float* __restrict__ Wb,
    const float* __restrict__ Wu, const float* __restrict__ Wv,
    float* __restrict__ ya, float* __restrict__ yb,
    float* __restrict__ yu, float* __restrict__ yv) {
  __shared__ float wlds[CCH * WP];
  const int w = blockIdx.y;
  const float* W = (w == 0) ? Wa : (w == 1) ? Wb : (w == 2) ? Wu : Wv;
  float*       Y = (w == 0) ? ya : (w == 1) ? yb : (w == 2) ? yu : yv;
  stage_weight(wlds, W);

  const int wave = threadIdx.x >> 5;
  const int lane = threadIdx.x & 31;
  const int lm = lane & 15, half = lane >> 4;
  const int rt = wave >> 1;
  const int n0 = (wave & 1) * 64;
  const int row0 = (blockIdx.x * 4 + rt) * 16;

  Strip s;
  wmma_strip(x + (size_t)row0 * CCH, wlds, n0, lm, half, s);

#pragma unroll
  for (int i = 0; i < 8; ++i) {
    size_t r = (size_t)(row0 + i + 8 * half) * CCH + n0 + lm;
    Y[r +  0] = s.acc[0][i];
    Y[r + 16] = s.acc[1][i];
    Y[r + 32] = s.acc[2][i];
    Y[r + 48] = s.acc[3][i];
  }
}

// ---------------------------------------------------------------------------
// Kernel 2: edge_upd = edge@E.T + xa[gather i] + xb[gather j]   (rows = B*144*T)
// block = 256 (8 waves): wave -> {row-tile 0..3} x {col-block 0..1}; E staged in LDS.
// ---------------------------------------------------------------------------
__global__ __launch_bounds__(256) void k_edgegemm(
    const float* __restrict__ edge, const float* __restrict__ We,
    const float* __restrict__ xa, const float* __restrict__ xb,
    float* __restrict__ eup) {
  __shared__ float wlds[CCH * WP];
  stage_weight(wlds, We);

  const int wave = threadIdx.x >> 5;
  const int lane = threadIdx.x & 31;
  const int lm = lane & 15, half = lane >> 4;
  const int rt = wave >> 1;
  const int n0 = (wave & 1) * 64;
  const int row0 = (blockIdx.x * 4 + rt) * 16;

  // 16 | TT, so a row tile has a single (b, e): decode gather indices once
  const int bidx = row0 / (NE * TT);
  const int rem  = row0 % (NE * TT);
  const int e    = rem / TT;
  const int t0   = rem % TT;
  const int i_n  = e / NNODE, j_n = e % NNODE;

  Strip s;
  wmma_strip(edge + (size_t)row0 * CCH, wlds, n0, lm, half, s);

  const float* xav = xa + ((size_t)(bidx * NNODE + i_n) * TT + t0) * CCH + n0 + lm;
  const float* xbv = xb + ((size_t)(bidx * NNODE + j_n) * TT + t0) * CCH + n0 + lm;
#pragma unroll
  for (int i = 0; i < 8; ++i) {
    const int ri = i + 8 * half;
    size_t ro = (size_t)(row0 + ri) * CCH + n0 + lm;
    size_t go = (size_t)ri * CCH;
    eup[ro +  0] = s.acc[0][i] + xav[go +  0] + xbv[go +  0];
    eup[ro + 16] = s.acc[1][i] + xav[go + 16] + xbv[go + 16];
    eup[ro + 32] = s.acc[2][i] + xav[go + 32] + xbv[go + 32];
    eup[ro + 48] = s.acc[3][i] + xav[go + 48] + xbv[go + 48];
  }
}

// ---------------------------------------------------------------------------
// Kernel 3: BatchNorm stats. One block per channel; stats over B*T*C elements.
// Layout of buf: (B, nchan, T*C). Biased variance, matching jnp.var.
// ---------------------------------------------------------------------------
__global__ __launch_bounds__(256) void k_bnstats(
    const float* __restrict__ buf, int nchan,
    float* __restrict__ mean, float* __restrict__ istd) {
  const int ch = blockIdx.x;
  const int M = BB * TC;
  float s = 0.f, s2 = 0.f;
  for (int u = threadIdx.x; u < M; u += 256) {
    int b = u >> 14;
    int off = u & (TC - 1);
    float v = buf[(size_t)(b * nchan + ch) * TC + off];
    s += v; s2 += v * v;
  }
  __shared__ float ss[256], ss2[256];
  ss[threadIdx.x] = s; ss2[threadIdx.x] = s2;
  __syncthreads();
  for (int st = 128; st > 0; st >>= 1) {
    if (threadIdx.x < st) {
      ss[threadIdx.x]  += ss[threadIdx.x + st];
      ss2[threadIdx.x] += ss2[threadIdx.x + st];
    }
    __syncthreads();
  }
  if (threadIdx.x == 0) {
    float m = ss[0] / (float)M;
    float var = ss2[0] / (float)M - m * m;
    mean[ch] = m;
    istd[ch] = rsqrtf(fmaxf(var, 0.f) + 1e-5f);
  }
}

// ---------------------------------------------------------------------------
// Kernel 4: edge_out = edge_in + relu(nanfix(bn(edge_upd)))   (float4 vectorized)
// ---------------------------------------------------------------------------
__global__ __launch_bounds__(256) void k_edge_update(
    const float* __restrict__ edge_in, const float* __restrict__ eup,
    const float* __restrict__ mean, const float* __restrict__ istd,
    const float* __restrict__ g, const float* __restrict__ bta,
    float* __restrict__ edge_out) {
  size_t i4 = (size_t)blockIdx.x * blockDim.x + threadIdx.x;
  if (i4 >= (size_t)SZE / 4) return;
  size_t idx = i4 * 4;
  int ch = (int)((idx >> 14) % NE);                 // TC = 2^14, chunks never straddle
  float m = mean[ch], is = istd[ch] * g[ch], bb = bta[ch];
  float4 u = ((const float4*)edge_in)[i4];
  float4 v = ((const float4*)eup)[i4];
  float uu[4] = {u.x, u.y, u.z, u.w};
  float vv[4] = {v.x, v.y, v.z, v.w};
  float r[4];
#pragma unroll
  for (int q = 0; q < 4; ++q) {
    float bn = (vv[q] - m) * is + bb;
    if (bn != bn) bn = 0.f;                         // NaN mask
    r[q] = uu[q] + fmaxf(bn, 0.f);
  }
  ((float4*)edge_out)[i4] = make_float4(r[0], r[1], r[2], r[3]);
}

// ---------------------------------------------------------------------------
// Kernel 5: fused sigmoid -> softmax(over j) -> weighted aggregate -> + x@U.T
// sigmoid in (0,1) => exp never overflows => no max-subtraction pass needed.
// tmp[b,i,t,c] = xu[b,i,t,c] + (sum_j e_j * xv[b,j,t,c]) / (sum_j e_j) / N
// ---------------------------------------------------------------------------
__global__ __launch_bounds__(256) void k_softmax_agg(
    const float* __restrict__ edge_new, const float* __restrict__ xv,
    const float* __restrict__ xu, float* __restrict__ tmp) {
  size_t idx = (size_t)blockIdx.x * blockDim.x + threadIdx.x;
  if (idx >= (size_t)SZX) return;
  int tc = (int)(idx & (TC - 1));
  int bi = (int)(idx >> 14);
  int b = bi / NNODE, i = bi % NNODE;
  const float* ep  = edge_new + ((size_t)(b * NE + i * NNODE)) * TC + tc;
  const float* xvp = xv + ((size_t)b * NNODE) * TC + tc;
  float denom = 0.f, num = 0.f;
#pragma unroll
  for (int j = 0; j < NNODE; ++j) {
    float v   = ep[(size_t)j * TC];
    float sig = 1.f / (1.f + expf(-v));
    float es  = expf(sig);
    denom += es;
    num   += es * xvp[(size_t)j * TC];
  }
  tmp[idx] = xu[idx] + num / (denom * (float)NNODE);
}

// ---------------------------------------------------------------------------
// Kernel 6: x_out = relu(x_in + nanfix?(bn(tmp)))   (float4 vectorized)
// ---------------------------------------------------------------------------
__global__ __launch_bounds__(256) void k_x_update(
    const float* __restrict__ x_in, const float* __restrict__ tmp,
    const float* __restrict__ mean, const float* __restrict__ istd,
    const float* __restrict__ g, const float* __restrict__ bta,
    int mask_nan, float* __restrict__ x_out) {
  size_t i4 = (size_t)blockIdx.x * blockDim.x + threadIdx.x;
  if (i4 >= (size_t)SZX / 4) return;
  size_t idx = i4 * 4;
  int ch = (int)((idx >> 14) % NNODE);
  float m = mean[ch], is = istd[ch] * g[ch], bb = bta[ch];
  float4 u = ((const float4*)x_in)[i4];
  float4 v = ((const float4*)tmp)[i4];
  float uu[4] = {u.x, u.y, u.z, u.w};
  float vv[4] = {v.x, v.y, v.z, v.w};
  float r[4];
#pragma unroll
  for (int q = 0; q < 4; ++q) {
    float bn = (vv[q] - m) * is + bb;
    if (mask_nan && bn != bn) bn = 0.f;
    r[q] = fmaxf(uu[q] + bn, 0.f);
  }
  ((float4*)x_out)[i4] = make_float4(r[0], r[1], r[2], r[3]);
}

// ---------------------------------------------------------------------------
// Host side
// ---------------------------------------------------------------------------
extern "C" void kernel_launch(void* const* d_in, const int* in_sizes, int n_in,
                              void* d_out, int out_size, void* d_ws, size_t ws_size,
                              hipStream_t stream) {
  (void)in_sizes; (void)n_in; (void)out_size; (void)ws_size;
  const float* x    = (const float*)d_in[0];
  const float* edge = (const float*)d_in[1];
  const float* Wt[10];
  for (int i = 0; i < 10; ++i) Wt[i] = (const float*)d_in[2 + i];
  // order: A1 B1 E1 U1 V1 A2 B2 E2 U2 V2
  const float* g_e1 = (const float*)d_in[12];
  const float* b_e1 = (const float*)d_in[13];
  const float* g_v1 = (const float*)d_in[14];
  const float* b_v1 = (const float*)d_in[15];
  const float* g_e2 = (const float*)d_in[16];
  const float* b_e2 = (const float*)d_in[17];
  const float* g_v2 = (const float*)d_in[18];
  const float* b_v2 = (const float*)d_in[19];

  float* x_out = (float*)d_out;           // (B,N,T,C)
  float* e_out = x_out + (size_t)SZX;     // (B,N*N,T,C)

  float* xa    = (float*)d_ws;            // also reused as "tmp" after edge gemm
  float* xb    = xa + (size_t)SZX;
  float* xu    = xb + (size_t)SZX;
  float* xv    = xu + (size_t)SZX;
  float* eup   = xv + (size_t)SZX;
  float* meanE = eup + (size_t)SZE;
  float* istdE = meanE + NE;
  float* meanV = istdE + NE;
  float* istdV = meanV + NNODE;

  const int BLK = 256;
  auto layer = [&](const float* xin, const float* ein,
                   const float* A, const float* Bw, const float* E,
                   const float* U, const float* V,
                   const float* ge, const float* be,
                   const float* gv, const float* bv, int mask) {
    k_xgemm<<<dim3(SZX / CCH / 64, 4), BLK, 0, stream>>>(xin, A, Bw, U, V, xa, xb, xu, xv);
    k_edgegemm<<<dim3(SZE / CCH / 64), BLK, 0, stream>>>(ein, E, xa, xb, eup);
    k_bnstats<<<dim3(NE), BLK, 0, stream>>>(eup, NE, meanE, istdE);
    k_edge_update<<<dim3(SZE / 4 / BLK), BLK, 0, stream>>>(ein, eup, meanE, istdE, ge, be, e_out);
    k_softmax_agg<<<dim3(SZX / BLK), BLK, 0, stream>>>(e_out, xv, xu, xa);
    k_bnstats<<<dim3(NNODE), BLK, 0, stream>>>(xa, NNODE, meanV, istdV);
    k_x_update<<<dim3(SZX / 4 / BLK), BLK, 0, stream>>>(xin, xa, meanV, istdV, gv, bv, mask, x_out);
  };

  // Layer 1: reads d_in x/edge, writes d_out x/edge slots
  layer(x, edge, Wt[0], Wt[1], Wt[2], Wt[3], Wt[4], g_e1, b_e1, g_v1, b_v1, 1);
  // Layer 2: in-place on d_out slots (all updates are elementwise residuals)
  layer(x_out, e_out, Wt[5], Wt[6], Wt[7], Wt[8], Wt[9], g_e2, b_e2, g_v2, b_v2, 0);
}